// GIN_v2_23055384445758
// MI455X (gfx1250) — compile-verified
//
#include <hip/hip_runtime.h>
#include <hip/hip_bf16.h>

typedef __attribute__((ext_vector_type(2))) float v2f;
typedef __attribute__((ext_vector_type(8))) float v8f;

#define F_DIM 128
#define H_DIM 256
#define C_DIM 40
#define CP_DIM 48   // C padded to 3 WMMA N-tiles

// ---------------- utility: zero fill ----------------
__global__ void zero_f32(float* __restrict__ p, int n) {
    int i = blockIdx.x * blockDim.x + threadIdx.x;
    if (i < n) p[i] = 0.f;
}

// ---------------- edge scatter-add: agg[dst] += x[src] ----------------
// 32 threads per edge, each handles 4 consecutive floats (float4 gather + 4 f32 atomics).
__global__ void __launch_bounds__(256)
scatter_kernel(const float* __restrict__ x, const int* __restrict__ ei,
               float* __restrict__ agg, int E) {
    long long tid = (long long)blockIdx.x * blockDim.x + threadIdx.x;
    int e = (int)(tid >> 5);
    if (e >= E) return;
    int part = (int)(tid & 31);
    int src = ei[e];
    int dst = ei[E + e];
    const float4 v = *(const float4*)(x + (size_t)src * F_DIM + part * 4);
    float* a = agg + (size_t)dst * F_DIM + part * 4;
    atomicAdd(a + 0, v.x);
    atomicAdd(a + 1, v.y);
    atomicAdd(a + 2, v.z);
    atomicAdd(a + 3, v.w);
}

// ---------------- h = (1+eps)*x + agg  (in place on agg) ----------------
__global__ void finalize_h(const float* __restrict__ x, const float* __restrict__ epsp,
                           float* __restrict__ agg, int n4) {
    int i = blockIdx.x * blockDim.x + threadIdx.x;
    if (i >= n4) return;
    float c = 1.f + epsp[0];
    float4 xv = ((const float4*)x)[i];
    float4 av = ((float4*)agg)[i];
    av.x += c * xv.x; av.y += c * xv.y; av.z += c * xv.z; av.w += c * xv.w;
    ((float4*)agg)[i] = av;
}

// ---------------- GEMM1: h1 = relu(h @ W1 + b1), fp32 WMMA 16x16x4 ----------------
// One wave per 16x16 output tile. A: [N,128] row-major, B: W1 [128,256] row-major.
__global__ void __launch_bounds__(256)
gemm1_wmma(const float* __restrict__ A, const float* __restrict__ W1,
           const float* __restrict__ b1, float* __restrict__ h1, int N) {
    int lane = threadIdx.x & 31;
    int wid  = threadIdx.x >> 5;
    int tile = blockIdx.x * 8 + wid;
    int mt = (N + 15) >> 4;
    if (tile >= mt * (H_DIM / 16)) return;   // wave-uniform exit
    int tm = tile / (H_DIM / 16);
    int tn = tile % (H_DIM / 16);
    int half = lane >> 4;        // 0: K pair {0,1}; 1: K pair {2,3}
    int ln   = lane & 15;

    int rowA = tm * 16 + ln;     // A-frag: M = lane%16
    if (rowA >= N) rowA = N - 1; // clamp (keeps EXEC all-1s)
    int colB = tn * 16 + ln;     // B/C/D: N = lane%16
    const float* arow = A  + (size_t)rowA * F_DIM + half * 2;
    const float* bcol = W1 + (size_t)(half * 2) * H_DIM + colB;

    v8f acc = {};
    for (int k = 0; k < F_DIM; k += 4) {
        v2f a = *(const v2f*)(arow + k);               // {K=k+2h, K=k+2h+1}
        v2f b;
        b.x = bcol[(size_t)k * H_DIM];                 // B[k+2h , n]
        b.y = bcol[(size_t)(k + 1) * H_DIM];           // B[k+2h+1, n]
        acc = __builtin_amdgcn_wmma_f32_16x16x4_f32(
            false, a, false, b, (short)0, acc, false, false);
    }

    float bb = b1[colB];
    int baseRow = tm * 16 + half * 8;                  // VGPR r -> M = r + 8*half
    for (int r = 0; r < 8; ++r) {
        int row = baseRow + r;
        float v = fmaxf(acc[r] + bb, 0.f);             // ReLU (act_first)
        if (row < N) h1[(size_t)row * H_DIM + colB] = v;
    }
}

// ---------------- BN stats: per-column sum / sumsq (coalesced partials) ----------------
__global__ void __launch_bounds__(256)
bn_partial(const float* __restrict__ h1, float* __restrict__ colsum,
           float* __restrict__ colsq, int N, int rpb) {
    int c  = threadIdx.x;
    int r0 = blockIdx.x * rpb;
    float s = 0.f, q = 0.f;
    for (int i = 0; i < rpb; ++i) {
        int r = r0 + i;
        if (r < N) {
            float v = h1[(size_t)r * H_DIM + c];
            s += v; q += v * v;
        }
    }
    atomicAdd(&colsum[c], s);
    atomicAdd(&colsq[c], q);
}

__global__ void bn_finalize(const float* __restrict__ colsum, const float* __restrict__ colsq,
                            const float* __restrict__ gamma,
                            float* __restrict__ meanb, float* __restrict__ sbuf, int N) {
    int c = threadIdx.x;
    float inv_n = 1.f / (float)N;
    float mean = colsum[c] * inv_n;
    float var  = colsq[c] * inv_n - mean * mean;
    meanb[c] = mean;
    sbuf[c]  = gamma[c] * rsqrtf(var + 1e-5f);
}

// ---------------- Fold BN into GEMM2 operands ----------------
// W2p[k, n(0..47)] = s[k]*W2[k,n] (0 pad for n>=40)
// biasp[n] = b2[n] + sum_k (beta[k] - mean[k]*s[k]) * W2[k,n]
__global__ void __launch_bounds__(256)
prep_w2(const float* __restrict__ W2, const float* __restrict__ b2,
        const float* __restrict__ beta, const float* __restrict__ meanb,
        const float* __restrict__ sbuf, float* __restrict__ W2p,
        float* __restrict__ biasp) {
    __shared__ float red[256];
    int n = blockIdx.x;     // 0..47
    int k = threadIdx.x;    // 0..255
    float w = (n < C_DIM) ? W2[k * C_DIM + n] : 0.f;
    float s = sbuf[k];
    W2p[k * CP_DIM + n] = s * w;
    red[k] = (beta[k] - meanb[k] * s) * w;
    __syncthreads();
    for (int d = 128; d > 0; d >>= 1) {
        if (k < d) red[k] += red[k + d];
        __syncthreads();
    }
    if (k == 0) biasp[n] = ((n < C_DIM) ? b2[n] : 0.f) + red[0];
}

// ---------------- GEMM2: out = h1 @ W2p + biasp, fp32 WMMA 16x16x4 ----------------
__global__ void __launch_bounds__(256)
gemm2_wmma(const float* __restrict__ A, const float* __restrict__ W2p,
           const float* __restrict__ biasp, float* __restrict__ out, int N) {
    int lane = threadIdx.x & 31;
    int wid  = threadIdx.x >> 5;
    int tile = blockIdx.x * 8 + wid;
    int mt = (N + 15) >> 4;
    if (tile >= mt * (CP_DIM / 16)) return;   // wave-uniform exit
    int tm = tile / (CP_DIM / 16);
    int tn = tile % (CP_DIM / 16);
    int half = lane >> 4;
    int ln   = lane & 15;

    int rowA = tm * 16 + ln;
    if (rowA >= N) rowA = N - 1;
    int colB = tn * 16 + ln;
    const float* arow = A   + (size_t)rowA * H_DIM + half * 2;
    const float* bcol = W2p + (size_t)(half * 2) * CP_DIM + colB;

    v8f acc = {};
    for (int k = 0; k < H_DIM; k += 4) {
        v2f a = *(const v2f*)(arow + k);
        v2f b;
        b.x = bcol[(size_t)k * CP_DIM];
        b.y = bcol[(size_t)(k + 1) * CP_DIM];
        acc = __builtin_amdgcn_wmma_f32_16x16x4_f32(
            false, a, false, b, (short)0, acc, false, false);
    }

    float bb = biasp[colB];
    int baseRow = tm * 16 + half * 8;
    for (int r = 0; r < 8; ++r) {
        int row = baseRow + r;
        if (row < N && colB < C_DIM)
            out[(size_t)row * C_DIM + colB] = acc[r] + bb;
    }
}

// ---------------- row-wise log_softmax over 40 cols, one wave32 per row ----------------
__global__ void __launch_bounds__(256)
logsoftmax_kernel(float* __restrict__ out, int N) {
    int lane = threadIdx.x & 31;
    int wid  = threadIdx.x >> 5;
    int row  = blockIdx.x * 8 + wid;
    if (row >= N) return;
    float* p = out + (size_t)row * C_DIM;
    float v1 = p[lane];                                        // cols 0..31
    bool has2 = lane < (C_DIM - 32);
    float v2 = has2 ? p[32 + lane] : -__builtin_inff();        // cols 32..39
    float m = fmaxf(v1, v2);
    for (int d = 16; d >= 1; d >>= 1) m = fmaxf(m, __shfl_xor(m, d, 32));
    float s = expf(v1 - m) + (has2 ? expf(v2 - m) : 0.f);
    for (int d = 16; d >= 1; d >>= 1) s += __shfl_xor(s, d, 32);
    float l = m + logf(s);
    p[lane] = v1 - l;
    if (has2) p[32 + lane] = v2 - l;
}

// ---------------- launcher ----------------
extern "C" void kernel_launch(void* const* d_in, const int* in_sizes, int n_in,
                              void* d_out, int out_size, void* d_ws, size_t ws_size,
                              hipStream_t stream) {
    const float* x     = (const float*)d_in[0];
    const int*   ei    = (const int*)d_in[1];
    const float* epsp  = (const float*)d_in[2];
    const float* W1    = (const float*)d_in[3];
    const float* b1    = (const float*)d_in[4];
    const float* gamma = (const float*)d_in[5];
    const float* beta  = (const float*)d_in[6];
    const float* W2    = (const float*)d_in[7];
    const float* b2    = (const float*)d_in[8];
    float* out = (float*)d_out;

    const int N = in_sizes[0] / F_DIM;   // 50000
    const int E = in_sizes[1] / 2;       // 800000

    float* ws     = (float*)d_ws;
    float* agg    = ws;                               // N*F  (becomes h)
    float* h1     = agg + (size_t)N * F_DIM;          // N*H
    float* colsum = h1 + (size_t)N * H_DIM;           // 256
    float* colsq  = colsum + H_DIM;                   // 256
    float* meanb  = colsq + H_DIM;                    // 256
    float* sbuf   = meanb + H_DIM;                    // 256
    float* W2p    = sbuf + H_DIM;                     // 256*48
    float* biasp  = W2p + H_DIM * CP_DIM;             // 48

    // zero agg + stats accumulators
    int nz = N * F_DIM;
    zero_f32<<<(nz + 255) / 256, 256, 0, stream>>>(agg, nz);
    zero_f32<<<2, 256, 0, stream>>>(colsum, 2 * H_DIM);

    // edge scatter-add (L2-resident gather + f32 atomics)
    long long st = (long long)E * 32;
    scatter_kernel<<<(int)((st + 255) / 256), 256, 0, stream>>>(x, ei, agg, E);

    // h = (1+eps)*x + agg
    int n4 = N * F_DIM / 4;
    finalize_h<<<(n4 + 255) / 256, 256, 0, stream>>>(x, epsp, agg, n4);

    // GEMM1 + bias + ReLU (WMMA fp32)
    int mt = (N + 15) / 16;
    int tiles1 = mt * (H_DIM / 16);
    gemm1_wmma<<<(tiles1 + 7) / 8, 256, 0, stream>>>(agg, W1, b1, h1, N);

    // BatchNorm stats + fold into GEMM2 operands
    int rpb = (N + 255) / 256;
    bn_partial<<<256, 256, 0, stream>>>(h1, colsum, colsq, N, rpb);
    bn_finalize<<<1, H_DIM, 0, stream>>>(colsum, colsq, gamma, meanb, sbuf, N);
    prep_w2<<<CP_DIM, 256, 0, stream>>>(W2, b2, beta, meanb, sbuf, W2p, biasp);

    // GEMM2 (+folded BN + bias) -> logits in d_out (WMMA fp32)
    int tiles2 = mt * (CP_DIM / 16);
    gemm2_wmma<<<(tiles2 + 7) / 8, 256, 0, stream>>>(h1, W2p, biasp, out, N);

    // log_softmax in place on d_out
    logsoftmax_kernel<<<(N + 7) / 8, 256, 0, stream>>>(out, N);
}